// StandardROIHeads_46351287058872
// MI455X (gfx1250) — compile-verified
//
#include <hip/hip_runtime.h>
#include <hip/hip_bf16.h>
#include <math.h>

#define B_IMGS 16
#define NPROP 4096
#define NCLS 80
#define CP1 81
#define MGT 64
#define PRE 1024
#define NDETS 100
#define NC (NPROP * NCLS) /* 327680 */
#define SCORE_TH 0.05f
#define NMS_TH 0.5f
#define MATCH_BG 0.5f
#define MATCH_FG 0.5f
#define SCALE_CLAMP 4.135166556742356f

typedef __attribute__((ext_vector_type(2))) float v2f;
typedef __attribute__((ext_vector_type(8))) float v8f;

__device__ __forceinline__ float iou_pp(float4 a, float4 b) {
  float areaA = (a.z - a.x + 1.f) * (a.w - a.y + 1.f);
  float areaB = (b.z - b.x + 1.f) * (b.w - b.y + 1.f);
  float lx = fmaxf(a.x, b.x), ly = fmaxf(a.y, b.y);
  float rx = fminf(a.z, b.z), ry = fminf(a.w, b.w);
  float w = fmaxf(rx - lx + 1.f, 0.f), h = fmaxf(ry - ly + 1.f, 0.f);
  float inter = w * h;
  return inter / (areaA + areaB - inter);
}

// ---------------- Matcher labels: [B,N] ----------------
__global__ void k_match(const float* __restrict__ props,
                        const float* __restrict__ gtb,
                        const int* __restrict__ gtl,
                        float* __restrict__ labels_out) {
  int b = blockIdx.y;
  int n = blockIdx.x * blockDim.x + threadIdx.x;
  __shared__ float4 sg[MGT];
  __shared__ int sl[MGT];
  if (threadIdx.x < MGT) {
    const float* g = gtb + (size_t)(b * MGT + threadIdx.x) * 4;
    sg[threadIdx.x] = make_float4(g[0], g[1], g[2], g[3]);
    sl[threadIdx.x] = gtl[b * MGT + threadIdx.x];
  }
  __syncthreads();
  const float* p = props + (size_t)(b * NPROP + n) * 4;
  float4 pb = make_float4(p[0], p[1], p[2], p[3]);
  float best = -1.f; int bi = 0;
  for (int m = 0; m < MGT; ++m) {
    float v = iou_pp(sg[m], pb);
    if (v > best) { best = v; bi = m; }  // first max wins (strict >)
  }
  int lbl = (best < MATCH_BG) ? 0 : ((best < MATCH_FG) ? -1 : sl[bi]);
  labels_out[b * NPROP + n] = (float)lbl;
}

// ---------------- Softmax fg scores: ws.scores[b][n*80+c] ----------------
__global__ void k_scores(const float* __restrict__ logits,
                         float* __restrict__ scores) {
  int b = blockIdx.y;
  int n = blockIdx.x * blockDim.x + threadIdx.x;
  const float* L = logits + (size_t)(b * NPROP + n) * CP1;
  float mx = L[0];
  for (int c = 1; c < CP1; ++c) mx = fmaxf(mx, L[c]);
  float s = 0.f;
  for (int c = 0; c < CP1; ++c) s += expf(L[c] - mx);
  float inv = 1.f / s;
  float* o = scores + (size_t)b * NC + (size_t)n * NCLS;
  for (int c = 1; c < CP1; ++c) o[c - 1] = expf(L[c] - mx) * inv;
}

// ---------------- MSB-first radix select of top-1024 per image ----------------
__global__ void __launch_bounds__(1024) k_select(const float* __restrict__ scores,
                                                 float* __restrict__ topv,
                                                 unsigned* __restrict__ topi) {
  int b = blockIdx.x;
  const float* S = scores + (size_t)b * NC;
  __shared__ unsigned hist[256];
  __shared__ unsigned s_prefVal, s_prefMask, s_k, s_G, s_cntG, s_cntT;
  if (threadIdx.x == 0) { s_prefVal = 0u; s_prefMask = 0u; s_k = PRE; }
  __syncthreads();
  for (int pass = 0; pass < 4; ++pass) {
    int shift = 24 - 8 * pass;
    if (threadIdx.x < 256) hist[threadIdx.x] = 0u;
    __syncthreads();
    unsigned pv = s_prefVal, pm = s_prefMask;
    for (int i = threadIdx.x; i < NC; i += 1024) {
      unsigned key = __float_as_uint(S[i]);   // probs > 0 => uint order == float order
      if ((key & pm) == pv) atomicAdd(&hist[(key >> shift) & 0xFFu], 1u);
    }
    __syncthreads();
    if (threadIdx.x == 0) {
      unsigned k = s_k, cum = 0u; int sel = 0;
      for (int bin = 255; bin >= 0; --bin) {
        unsigned c = hist[bin];
        if (cum + c >= k) { sel = bin; break; }
        cum += c;
      }
      s_k = k - cum;
      s_prefVal = pv | ((unsigned)sel << shift);
      s_prefMask = pm | (0xFFu << shift);
    }
    __syncthreads();
  }
  unsigned T = s_prefVal;
  if (threadIdx.x == 0) { s_G = PRE - s_k; s_cntG = 0u; s_cntT = 0u; }
  __syncthreads();
  unsigned G = s_G, kT = s_k;
  float* tv = topv + b * PRE;
  unsigned* ti = topi + b * PRE;
  for (int i = threadIdx.x; i < NC; i += 1024) {
    float v = S[i];
    unsigned key = __float_as_uint(v);
    if (key > T) {
      unsigned pos = atomicAdd(&s_cntG, 1u);
      tv[pos] = v; ti[pos] = (unsigned)i;
    } else if (key == T) {
      unsigned t = atomicAdd(&s_cntT, 1u);
      if (t < kT) { tv[G + t] = v; ti[G + t] = (unsigned)i; }
    }
  }
}

// ------------- bitonic sort 1024 (desc score, asc idx) + box decode -------------
__global__ void __launch_bounds__(1024) k_sort_decode(
    float* __restrict__ topv, unsigned* __restrict__ topi,
    const float* __restrict__ props, const float* __restrict__ reg,
    const int* __restrict__ ih, const int* __restrict__ iw,
    float* __restrict__ tb, float* __restrict__ sbx,
    float* __restrict__ areaS, unsigned* __restrict__ clsArr) {
  int b = blockIdx.x;
  unsigned tid = threadIdx.x;
  __shared__ float sv[PRE];
  __shared__ unsigned si[PRE];
  sv[tid] = topv[b * PRE + tid];
  si[tid] = topi[b * PRE + tid];
  __syncthreads();
  for (unsigned k = 2; k <= PRE; k <<= 1) {
    for (unsigned j = k >> 1; j > 0; j >>= 1) {
      unsigned ixj = tid ^ j;
      if (ixj > tid) {
        float va = sv[tid], vb = sv[ixj];
        unsigned ia = si[tid], ib = si[ixj];
        bool a_after_b = (va < vb) || (va == vb && ia > ib);
        bool descSeg = ((tid & k) == 0u);
        bool doSwap = descSeg ? a_after_b : !a_after_b;
        if (doSwap) { sv[tid] = vb; sv[ixj] = va; si[tid] = ib; si[ixj] = ia; }
      }
      __syncthreads();
    }
  }
  topv[b * PRE + tid] = sv[tid];
  topi[b * PRE + tid] = si[tid];
  unsigned flat = si[tid];
  unsigned n = flat / NCLS, c0 = flat % NCLS;     // fg class 0..79, real class c0+1
  const float* p = props + (size_t)(b * NPROP + n) * 4;
  float x1 = p[0], y1 = p[1], x2 = p[2], y2 = p[3];
  float w = x2 - x1 + 1.f, h = y2 - y1 + 1.f;
  float cx = x1 + 0.5f * w, cy = y1 + 0.5f * h;
  const float* d = reg + (size_t)(b * NPROP + n) * (CP1 * 4) + (size_t)(c0 + 1) * 4;
  float dx = d[0] * 0.1f, dy = d[1] * 0.1f;               // / (10,10)
  float dw = fminf(d[2] * 0.2f, SCALE_CLAMP);             // / 5, clamp
  float dh = fminf(d[3] * 0.2f, SCALE_CLAMP);
  float W = (float)iw[0], H = (float)ih[0];
  float pcx = dx * w + cx, pcy = dy * h + cy;
  float pw = expf(dw) * w, ph = expf(dh) * h;
  float bx1 = fminf(fmaxf(pcx - 0.5f * pw, 0.f), W - 1.f);
  float by1 = fminf(fmaxf(pcy - 0.5f * ph, 0.f), H - 1.f);
  float bx2 = fminf(fmaxf(pcx + 0.5f * pw - 1.f, 0.f), W - 1.f);
  float by2 = fminf(fmaxf(pcy + 0.5f * ph - 1.f, 0.f), H - 1.f);
  float* t = tb + (size_t)(b * PRE + tid) * 4;
  t[0] = bx1; t[1] = by1; t[2] = bx2; t[3] = by2;
  float off = (float)c0 * (2.f * (W + H));                // class-offset trick
  float* s = sbx + (size_t)(b * PRE + tid) * 4;
  s[0] = bx1 + off; s[1] = by1 + off; s[2] = bx2 + off; s[3] = by2 + off;
  areaS[b * PRE + tid] = (bx2 - bx1 + 1.f) * (by2 - by1 + 1.f); // shift cancels
  clsArr[b * PRE + tid] = c0;
}

// ---- 16x32 IoU tiles: union areas via two V_WMMA_F32_16X16X4_F32 (shared A) ----
// Each wave owns full 32-bit words of the suppression bit matrix: no atomics,
// no zero-init. Blocks entirely below the diagonal (never consulted, since the
// greedy pass only uses bits j > i) write zeros and exit early.
__global__ void __launch_bounds__(32) k_ioubits(const float* __restrict__ sbx,
                                                const float* __restrict__ areaS,
                                                unsigned* __restrict__ bits) {
  int b = blockIdx.z;
  int tjw = blockIdx.x;          // 32-column word tile, 0..31
  int ti = blockIdx.y;           // 16-row tile, 0..63
  int lane = threadIdx.x;
  __shared__ unsigned sword[16];
  unsigned* wout = bits + ((size_t)b * PRE + (size_t)ti * 16) * 32 + tjw;
  if (tjw * 32 + 31 <= ti * 16) {          // all columns <= min row: bits unused
    if (lane < 16) wout[(size_t)lane * 32] = 0u;
    return;
  }
  int nLoc = lane & 15;
  int half = lane >> 4;                    // 0: lanes 0-15, 1: lanes 16-31
  int col0 = tjw * 32 + nLoc;
  int col1 = col0 + 16;
  const float4 cb0 = *(const float4*)(sbx + (size_t)(b * PRE + col0) * 4);
  const float4 cb1 = *(const float4*)(sbx + (size_t)(b * PRE + col1) * 4);
  float areaC0 = areaS[b * PRE + col0];
  float areaC1 = areaS[b * PRE + col1];
  float areaR = areaS[b * PRE + ti * 16 + nLoc];   // A layout: M = lane&15
  v2f a, b0, b1;
  a.x = half ? 0.f : areaR;   a.y = half ? 0.f : 1.f;    // A: K0=area, K1=1
  b0.x = half ? 0.f : 1.f;    b0.y = half ? 0.f : areaC0; // B: K0=1s, K1=areaB
  b1.x = half ? 0.f : 1.f;    b1.y = half ? 0.f : areaC1;
  v8f c0 = {}, c1 = {};
  c0 = __builtin_amdgcn_wmma_f32_16x16x4_f32(false, a, false, b0,
                                             (short)0, c0, false, false);
  c1 = __builtin_amdgcn_wmma_f32_16x16x4_f32(false, a, false, b1,
                                             (short)0, c1, false, false);
  for (int v = 0; v < 8; ++v) {
    int row = ti * 16 + v + half * 8;      // C layout row for this lane/VGPR
    const float4 rb = *(const float4*)(sbx + (size_t)(b * PRE + row) * 4);
    float lx0 = fmaxf(rb.x, cb0.x), ly0 = fmaxf(rb.y, cb0.y);
    float rx0 = fminf(rb.z, cb0.z), ry0 = fminf(rb.w, cb0.w);
    float in0 = fmaxf(rx0 - lx0 + 1.f, 0.f) * fmaxf(ry0 - ly0 + 1.f, 0.f);
    float lx1 = fmaxf(rb.x, cb1.x), ly1 = fmaxf(rb.y, cb1.y);
    float rx1 = fminf(rb.z, cb1.z), ry1 = fminf(rb.w, cb1.w);
    float in1 = fmaxf(rx1 - lx1 + 1.f, 0.f) * fmaxf(ry1 - ly1 + 1.f, 0.f);
    float iou0 = in0 / (c0[v] - in0);      // union = areaSum - inter
    float iou1 = in1 / (c1[v] - in1);
    unsigned m0 = __builtin_amdgcn_ballot_w32(iou0 > NMS_TH);
    unsigned m1 = __builtin_amdgcn_ballot_w32(iou1 > NMS_TH);
    if (lane == v) {                       // masks are wave-uniform
      sword[v]     = (m0 & 0xFFFFu) | ((m1 & 0xFFFFu) << 16);  // row M=v
      sword[v + 8] = (m0 >> 16)     | (m1 & 0xFFFF0000u);      // row M=v+8
    }
  }
  __syncthreads();
  if (lane < 16) wout[(size_t)lane * 32] = sword[lane];
}

// ---------------- serial greedy NMS, single wave32 per image ----------------
__global__ void __launch_bounds__(32) k_nms_out(const float* __restrict__ topv,
                                                const unsigned* __restrict__ bits,
                                                const float* __restrict__ tb,
                                                const unsigned* __restrict__ clsArr,
                                                float* __restrict__ dets) {
  int b = blockIdx.x;
  int lane = threadIdx.x;
  __shared__ unsigned keep_idx[NDETS];
  unsigned sup = 0u;                      // suppression word owned by this lane
  int kept = 0;                           // uniform across wave
  const unsigned* Bits = bits + (size_t)b * PRE * 32;
  const float* tv = topv + b * PRE;
  for (int i = 0; i < PRE; ++i) {
    unsigned supWord = (unsigned)__shfl((int)sup, i >> 5, 32);
    bool suppressed = (supWord >> (i & 31)) & 1u;
    if (!suppressed) {
      float sv_ = tv[i];
      if (sv_ > SCORE_TH && kept < NDETS) {
        if (lane == 0) keep_idx[kept] = (unsigned)i;
        kept++;
      }
      unsigned row = Bits[(size_t)i * 32 + lane];
      int wi = i >> 5, bi = i & 31;
      unsigned m;                         // keep only bits j > i
      if (lane < wi) m = 0u;
      else if (lane == wi) m = (bi == 31) ? 0u : (~0u << (bi + 1));
      else m = ~0u;
      sup |= row & m;
    }
  }
  __syncthreads();
  float* D = dets + (size_t)b * NDETS * 6;
  for (int t = lane; t < NDETS; t += 32) {
    if (t < kept) {
      unsigned idx = keep_idx[t];
      const float* bx = tb + (size_t)(b * PRE + idx) * 4;
      D[t * 6 + 0] = bx[0]; D[t * 6 + 1] = bx[1];
      D[t * 6 + 2] = bx[2]; D[t * 6 + 3] = bx[3];
      D[t * 6 + 4] = tv[idx];
      D[t * 6 + 5] = (float)(clsArr[b * PRE + idx] + 1u);
    } else {
      for (int f = 0; f < 6; ++f) D[t * 6 + f] = 0.f;
    }
  }
}

extern "C" void kernel_launch(void* const* d_in, const int* in_sizes, int n_in,
                              void* d_out, int out_size, void* d_ws, size_t ws_size,
                              hipStream_t stream) {
  (void)in_sizes; (void)n_in; (void)out_size; (void)ws_size;
  const float* props  = (const float*)d_in[0];
  const float* logits = (const float*)d_in[1];
  const float* reg    = (const float*)d_in[2];
  const float* gtb    = (const float*)d_in[3];
  const int*   gtl    = (const int*)d_in[4];
  const int*   ih     = (const int*)d_in[5];
  const int*   iw     = (const int*)d_in[6];

  float* dets   = (float*)d_out;                       // [B,100,6]
  float* labels = dets + (size_t)B_IMGS * NDETS * 6;   // [B,4096]

  char* ws = (char*)d_ws;
  float*    scores = (float*)ws;    ws += sizeof(float)    * (size_t)B_IMGS * NC;
  float*    topv   = (float*)ws;    ws += sizeof(float)    * (size_t)B_IMGS * PRE;
  unsigned* topi   = (unsigned*)ws; ws += sizeof(unsigned) * (size_t)B_IMGS * PRE;
  float*    tb     = (float*)ws;    ws += sizeof(float)    * (size_t)B_IMGS * PRE * 4;
  float*    sbx    = (float*)ws;    ws += sizeof(float)    * (size_t)B_IMGS * PRE * 4;
  float*    areaS  = (float*)ws;    ws += sizeof(float)    * (size_t)B_IMGS * PRE;
  unsigned* clsArr = (unsigned*)ws; ws += sizeof(unsigned) * (size_t)B_IMGS * PRE;
  unsigned* bits   = (unsigned*)ws; // [B][1024][32] words, fully written each call

  k_match<<<dim3(NPROP / 256, B_IMGS), 256, 0, stream>>>(props, gtb, gtl, labels);
  k_scores<<<dim3(NPROP / 256, B_IMGS), 256, 0, stream>>>(logits, scores);
  k_select<<<B_IMGS, 1024, 0, stream>>>(scores, topv, topi);
  k_sort_decode<<<B_IMGS, PRE, 0, stream>>>(topv, topi, props, reg, ih, iw,
                                            tb, sbx, areaS, clsArr);
  k_ioubits<<<dim3(PRE / 32, PRE / 16, B_IMGS), 32, 0, stream>>>(sbx, areaS, bits);
  k_nms_out<<<B_IMGS, 32, 0, stream>>>(topv, bits, tb, clsArr, dets);
}